// AdvancedGCN_17231408792366
// MI455X (gfx1250) — compile-verified
//
#include <hip/hip_runtime.h>
#include <hip/hip_bf16.h>
#include <stdint.h>

// ---------------------------------------------------------------------------
// AdvancedGCN on MI455X (gfx1250, wave32).
//  - Dense 128x128 projections: V_WMMA_F32_16X16X4_F32, 32 chained K=4 steps,
//    one 16x16 C tile per wave, A tile staged to LDS with
//    GLOBAL_LOAD_ASYNC_TO_LDS_B128 (ASYNCcnt + s_wait_asynccnt).
//  - Graph aggregate: agg initialized with the self-loop term (saves a
//    zero-fill pass and a combine pass), then one wave per edge does a float4
//    gather + 4x global_atomic_add_f32 scatter (all L2-resident: 51 MB
//    feature matrix << 192 MB L2). global_prefetch for future gather rows.
// ---------------------------------------------------------------------------

typedef float v2f __attribute__((ext_vector_type(2)));
typedef float v8f __attribute__((ext_vector_type(8)));

#define D 128
#define EPS_BN 1e-5f
#define EPS_LN 1e-5f
#define USE_ASYNC_LDS 1
#define SCATTER_PF 16   // edge-lookahead for gather-row prefetch

// ---------------------------------------------------------------------------
__global__ void k_zero_f32(float* __restrict__ p, long long n) {
    long long gid = (long long)blockIdx.x * blockDim.x + threadIdx.x;
    if (gid < n) p[gid] = 0.0f;
}

// degree accumulation: deg[col[e]] += 1
__global__ void k_deg(const int* __restrict__ cols, float* __restrict__ deg, int e) {
    long long gid = (long long)blockIdx.x * blockDim.x + threadIdx.x;
    if (gid < e) atomicAdd(&deg[cols[gid]], 1.0f);
}

// dinv[i] = rsqrt(deg[i] + 1)   (in place)
__global__ void k_dinv(float* __restrict__ d, int n) {
    long long gid = (long long)blockIdx.x * blockDim.x + threadIdx.x;
    if (gid < n) d[gid] = rsqrtf(d[gid] + 1.0f);
}

// ---------------------------------------------------------------------------
// C[n,128] = A[n,128] @ W[128,128] (+ bias per column if use_bias)
// Block: 256 threads = 8 waves; wave w -> 16x16 tile at columns [16w,16w+16).
//
// fp32 WMMA fragment layouts (ISA 7.12.2):
//   A (16x4): lanes 0-15 -> {K=k0,k0+1}, lanes 16-31 -> {K=k0+2,k0+3}, M=lane%16
//   B (4x16): mirrored K split per lane half, N=lane%16
//   C (16x16): acc[r] -> row r (lanes 0-15) / r+8 (lanes 16-31), col=lane%16
__global__ __launch_bounds__(256) void k_gemm128_wmma(
        const float* __restrict__ A, const float* __restrict__ W,
        const float* __restrict__ bias, float* __restrict__ C,
        int nrows, int use_bias) {
    __shared__ __align__(16) float As[16 * D];   // 8 KB row tile

    const int tid  = threadIdx.x;
    const int wave = tid >> 5;
    const int lane = tid & 31;
    const int row0 = blockIdx.x * 16;

    // stage the 16x128 A tile into LDS: 2 x 16B per thread
    for (int i = tid * 4; i < 16 * D; i += 256 * 4) {
        int tr = i >> 7;            // tile row 0..15
        int tc = i & (D - 1);       // col
        int gr = row0 + tr;
        if (gr >= nrows) gr = nrows - 1;           // clamp (duplicate read ok)
        const float* gsrc = &A[(size_t)gr * D + tc];
#if USE_ASYNC_LDS
        unsigned int lds_off = (unsigned int)(size_t)&As[i];  // low 32b = LDS byte addr
        unsigned long long gaddr = (unsigned long long)(size_t)gsrc;
        asm volatile("global_load_async_to_lds_b128 %0, %1, off"
                     :: "v"(lds_off), "v"(gaddr) : "memory");
#else
        *(float4*)&As[i] = *(const float4*)gsrc;
#endif
    }
#if USE_ASYNC_LDS
    asm volatile("s_wait_asynccnt 0x0" ::: "memory");  // drain this wave's ASYNCcnt
#endif
    __syncthreads();

    const int m    = lane & 15;               // row in A frag / col in B,C frags
    const int koff = (lane < 16) ? 0 : 2;     // K sub-offset per lane half
    const int colB = wave * 16;

    v8f acc = {};
    for (int k0 = 0; k0 < D; k0 += 4) {
        v2f a, b;
        a.x = As[m * D + k0 + koff];
        a.y = As[m * D + k0 + koff + 1];
        b.x = W[(size_t)(k0 + koff) * D + colB + m];
        b.y = W[(size_t)(k0 + koff + 1) * D + colB + m];
        acc = __builtin_amdgcn_wmma_f32_16x16x4_f32(
                  /*neg_a=*/false, a, /*neg_b=*/false, b,
                  /*c_mod=*/(short)0, acc, /*reuse_a=*/false, /*reuse_b=*/false);
    }

    const int mbase = (lane < 16) ? 0 : 8;
    const float bb  = use_bias ? bias[colB + m] : 0.0f;
#pragma unroll
    for (int r = 0; r < 8; ++r) {
        int gr = row0 + mbase + r;
        if (gr < nrows)
            C[(size_t)gr * D + colB + m] = acc[r] + bb;
    }
}

// ---------------------------------------------------------------------------
// agg[i] = h[i] * dinv[i]^2 + b[c]   (self-loop term + bias; scatter adds on top)
__global__ void k_selfinit(const float* __restrict__ h, const float* __restrict__ dinv,
                           const float* __restrict__ bias, float* __restrict__ out, int n) {
    long long gid = (long long)blockIdx.x * blockDim.x + threadIdx.x;
    int i    = (int)(gid >> 5);
    int lane = (int)(gid & 31);
    if (i >= n) return;
    float di = dinv[i];
    float sl = di * di;
    size_t base = (size_t)i * D + lane * 4;
    float4 hv = *(const float4*)&h[base];
    int c = lane * 4;
    float4 o;
    o.x = hv.x * sl + bias[c + 0];
    o.y = hv.y * sl + bias[c + 1];
    o.z = hv.z * sl + bias[c + 2];
    o.w = hv.w * sl + bias[c + 3];
    *(float4*)&out[base] = o;
}

// One wave per edge: agg[col] += h[row] * dinv[row]*dinv[col]
__global__ void k_scatter(const int* __restrict__ rows, const int* __restrict__ cols,
                          const float* __restrict__ dinv, const float* __restrict__ h,
                          float* __restrict__ agg, int e) {
    long long gid = (long long)blockIdx.x * blockDim.x + threadIdx.x;
    int edge = (int)(gid >> 5);
    int lane = (int)(gid & 31);
    if (edge >= e) return;
    // prefetch a future edge's gather row into cache (global_prefetch)
    int pfe = edge + SCATTER_PF;
    if (pfe < e) {
        int rpf = rows[pfe];
        __builtin_prefetch(&h[(size_t)rpf * D + lane * 4], 0, 0);
    }
    int r = rows[edge], c = cols[edge];
    float w = dinv[r] * dinv[c];
    const float4 v = *(const float4*)&h[(size_t)r * D + lane * 4];
    float* dst = &agg[(size_t)c * D + lane * 4];
    atomicAdd(dst + 0, v.x * w);
    atomicAdd(dst + 1, v.y * w);
    atomicAdd(dst + 2, v.z * w);
    atomicAdd(dst + 3, v.w * w);
}

// ---------------------------------------------------------------------------
// per-column sum / sumsq over N rows (stats[0..127]=sum, stats[128..255]=sumsq)
__global__ __launch_bounds__(256) void k_colstats(const float* __restrict__ X,
                                                  float* __restrict__ stats, int n) {
    int c  = threadIdx.x & (D - 1);
    int r0 = blockIdx.x * 2 + (threadIdx.x >> 7);
    int rs = gridDim.x * 2;
    float s = 0.0f, s2 = 0.0f;
    for (int r = r0; r < n; r += rs) {
        float v = X[(size_t)r * D + c];
        s += v; s2 += v * v;
    }
    atomicAdd(&stats[c], s);
    atomicAdd(&stats[D + c], s2);
}

// out = relu((conv - mu)*rsqrt(var+eps)*g + be) + res
__global__ void k_bn_relu_res(const float* __restrict__ conv, const float* __restrict__ res,
                              const float* __restrict__ stats,
                              const float* __restrict__ g, const float* __restrict__ be,
                              float* __restrict__ out, int n) {
    long long gid = (long long)blockIdx.x * blockDim.x + threadIdx.x;
    long long total = (long long)n * D;
    if (gid >= total) return;
    int c = (int)(gid & (D - 1));
    float invn = 1.0f / (float)n;
    float mu  = stats[c] * invn;
    float var = stats[D + c] * invn - mu * mu;
    float iv  = rsqrtf(var + EPS_BN);
    float v = (conv[gid] - mu) * iv * g[c] + be[c];
    v = v > 0.0f ? v : 0.0f;
    out[gid] = v + res[gid];
}

// ---------------------------------------------------------------------------
// out = LayerNorm(hin + att) over the 128-feature axis; one wave per row.
__global__ void k_add_layernorm(const float* __restrict__ hin, const float* __restrict__ att,
                                const float* __restrict__ g, const float* __restrict__ b,
                                float* __restrict__ out, int n) {
    long long gid = (long long)blockIdx.x * blockDim.x + threadIdx.x;
    int row  = (int)(gid >> 5);
    int lane = (int)(gid & 31);
    if (row >= n) return;
    size_t base = (size_t)row * D + lane * 4;
    float4 h = *(const float4*)&hin[base];
    float4 a = *(const float4*)&att[base];
    float4 t;
    t.x = h.x + a.x; t.y = h.y + a.y; t.z = h.z + a.z; t.w = h.w + a.w;
    float s  = t.x + t.y + t.z + t.w;
    float s2 = t.x * t.x + t.y * t.y + t.z * t.z + t.w * t.w;
#pragma unroll
    for (int m = 16; m >= 1; m >>= 1) {
        s  += __shfl_xor(s,  m, 32);
        s2 += __shfl_xor(s2, m, 32);
    }
    float mu  = s * (1.0f / D);
    float var = s2 * (1.0f / D) - mu * mu;
    float iv  = rsqrtf(var + EPS_LN);
    int c = lane * 4;
    float4 o;
    o.x = (t.x - mu) * iv * g[c + 0] + b[c + 0];
    o.y = (t.y - mu) * iv * g[c + 1] + b[c + 1];
    o.z = (t.z - mu) * iv * g[c + 2] + b[c + 2];
    o.w = (t.w - mu) * iv * g[c + 3] + b[c + 3];
    *(float4*)&out[base] = o;
}

// ---------------------------------------------------------------------------
extern "C" void kernel_launch(void* const* d_in, const int* in_sizes, int n_in,
                              void* d_out, int out_size, void* d_ws, size_t ws_size,
                              hipStream_t stream) {
    const float* x   = (const float*)d_in[0];
    const int*   ei  = (const int*)d_in[1];
    const float* W0  = (const float*)d_in[2];  const float* b0  = (const float*)d_in[3];
    const float* W1  = (const float*)d_in[4];  const float* b1  = (const float*)d_in[5];
    const float* W2  = (const float*)d_in[6];  const float* b2  = (const float*)d_in[7];
    const float* g0  = (const float*)d_in[8];  const float* be0 = (const float*)d_in[9];
    const float* g1  = (const float*)d_in[10]; const float* be1 = (const float*)d_in[11];
    const float* Wv  = (const float*)d_in[12]; const float* bv  = (const float*)d_in[13];
    const float* Wo  = (const float*)d_in[14]; const float* bo  = (const float*)d_in[15];
    const float* lng = (const float*)d_in[16]; const float* lnb = (const float*)d_in[17];

    const int n = in_sizes[0] / D;
    const int e = in_sizes[1] / 2;
    const int* rows = ei;       // edge_index[0]
    const int* cols = ei + e;   // edge_index[1]

    // ---- workspace carve (256-byte aligned) ----
    char* wsb = (char*)d_ws;
    size_t off = 0;
    auto carve = [&](size_t bytes) -> float* {
        float* p = (float*)(wsb + off);
        off = (off + bytes + 255) & ~(size_t)255;
        return p;
    };
    const size_t nd_bytes = (size_t)n * D * sizeof(float);
    float* dinv  = carve((size_t)n * sizeof(float));
    float* stats = carve(2 * D * sizeof(float));
    float* B0    = carve(nd_bytes);
    float* B1    = carve(nd_bytes);
    float* B2    = carve(nd_bytes);
    (void)ws_size; (void)n_in; (void)out_size;

    const int TB = 256;
    const long long nd   = (long long)n * D;
    const int blk_nd     = (int)((nd + TB - 1) / TB);
    const int blk_n      = (n + TB - 1) / TB;
    const int blk_e      = (e + TB - 1) / TB;
    const int blk_edge32 = (int)(((long long)e * 32 + TB - 1) / TB);
    const int blk_row32  = (int)(((long long)n * 32 + TB - 1) / TB);
    const int blk_gemm   = (n + 15) / 16;
    const int blk_stats  = 512;

    // ---- degrees (shared by all three convs) ----
    k_zero_f32<<<blk_n, TB, 0, stream>>>(dinv, n);
    k_deg<<<blk_e, TB, 0, stream>>>(cols, dinv, e);
    k_dinv<<<blk_n, TB, 0, stream>>>(dinv, n);

    // ---- conv layers 0 and 1: conv -> BN -> relu -> +res ----
    const float* Ws[2]  = {W0, W1};
    const float* bs[2]  = {b0, b1};
    const float* gs[2]  = {g0, g1};
    const float* bes[2] = {be0, be1};
    const float* cur = x;

    for (int l = 0; l < 2; ++l) {
        // l==0: cur=x,  h=B0, agg=B1, out=B2
        // l==1: cur=B2, h=B1, agg=B0, out=B0 (elementwise alias, safe)
        float* h   = (l == 0) ? B0 : B1;
        float* agg = (l == 0) ? B1 : B0;
        float* nxt = (l == 0) ? B2 : B0;

        k_gemm128_wmma<<<blk_gemm, TB, 0, stream>>>(cur, Ws[l], nullptr, h, n, 0);
        k_selfinit<<<blk_row32, TB, 0, stream>>>(h, dinv, bs[l], agg, n);
        k_scatter<<<blk_edge32, TB, 0, stream>>>(rows, cols, dinv, h, agg, e);
        k_zero_f32<<<1, TB, 0, stream>>>(stats, 2 * D);
        k_colstats<<<blk_stats, TB, 0, stream>>>(agg, stats, n);
        k_bn_relu_res<<<blk_nd, TB, 0, stream>>>(agg, cur, stats, gs[l], bes[l], nxt, n);
        cur = nxt;
    }
    // now: cur = B0 (layer1 output); B1, B2 free.

    // ---- MHA (seq_len==1): att = (cur@Wv + bv)@Wo + bo; h = LN(cur + att) ----
    k_gemm128_wmma<<<blk_gemm, TB, 0, stream>>>(cur, Wv, bv, B1, n, 1);
    k_gemm128_wmma<<<blk_gemm, TB, 0, stream>>>(B1, Wo, bo, B2, n, 1);
    k_add_layernorm<<<blk_row32, TB, 0, stream>>>(cur, B2, lng, lnb, B1, n);
    cur = B1;   // B0, B2 free

    // ---- output conv: self-loop init + scatter straight into d_out ----
    float* out = (float*)d_out;
    k_gemm128_wmma<<<blk_gemm, TB, 0, stream>>>(cur, W2, nullptr, B0, n, 0);
    k_selfinit<<<blk_row32, TB, 0, stream>>>(B0, dinv, b2, out, n);
    k_scatter<<<blk_edge32, TB, 0, stream>>>(rows, cols, dinv, B0, out, e);
}